// GaussianKmeans_77524159693493
// MI455X (gfx1250) — compile-verified
//
#include <hip/hip_runtime.h>

#define B_    32
#define S_    16384
#define D_    64
#define C_    10
#define NCH   16            // chunks per batch
#define CHUNK (S_ / NCH)    // 1024 rows per block
#define NBLK  (B_ * NCH)    // 512 partial blocks

typedef float v2f __attribute__((ext_vector_type(2)));
typedef float v8f __attribute__((ext_vector_type(8)));

// ---------------------------------------------------------------------------
// Kernel 1: per-(batch,chunk) segmented sum of x by cluster label.
// Deterministic: each thread owns a private LDS region (row-group x d-slice),
// final reductions run in a fixed order. No atomics.
// ---------------------------------------------------------------------------
__global__ void __launch_bounds__(256)
segsum_kernel(const float* __restrict__ x, const int* __restrict__ labels,
              float* __restrict__ psum, float* __restrict__ pcnt) {
    __shared__ float sums[16][C_ * D_];   // 40 KB
    __shared__ float cntp[16][C_];

    const int tid = threadIdx.x;
    const int rg  = tid >> 4;   // row-group 0..15
    const int dl  = tid & 15;   // d-slice (4 floats each)
    const int b   = blockIdx.y;

    for (int i = tid; i < 16 * C_ * D_; i += 256) ((float*)sums)[i] = 0.f;
    if (tid < 16 * C_) ((float*)cntp)[tid] = 0.f;
    __syncthreads();

    const float4* xv = (const float4*)x;
    const int s0 = blockIdx.x * CHUNK;
    for (int i = 0; i < CHUNK / 16; ++i) {
        const int s = s0 + i * 16 + rg;
        const int c = labels[b * S_ + s];
        const float4 v = xv[(b * S_ + s) * (D_ / 4) + dl];
        float4* acc = (float4*)&sums[rg][c * D_ + dl * 4];
        float4 o = *acc;
        o.x += v.x; o.y += v.y; o.z += v.z; o.w += v.w;
        *acc = o;
        if (dl == 0) cntp[rg][c] += 1.f;
    }
    __syncthreads();

    const int blk = b * NCH + blockIdx.x;
    for (int i = tid; i < C_ * D_; i += 256) {
        float a = 0.f;
#pragma unroll
        for (int r = 0; r < 16; ++r) a += sums[r][i];
        psum[blk * (C_ * D_) + i] = a;
    }
    if (tid < C_) {
        float a = 0.f;
#pragma unroll
        for (int r = 0; r < 16; ++r) a += cntp[r][tid];
        pcnt[blk * C_ + tid] = a;
    }
}

// ---------------------------------------------------------------------------
// Kernel 2: combine the 16 chunk-partials per batch, divide by counts -> cc.
// ---------------------------------------------------------------------------
__global__ void __launch_bounds__(256)
reduce_cc_kernel(const float* __restrict__ psum, const float* __restrict__ pcnt,
                 float* __restrict__ cc) {
    const int idx = blockIdx.x * 256 + threadIdx.x;   // [0, 320*64)
    if (idx >= B_ * C_ * D_) return;
    const int d  = idx & (D_ - 1);
    const int bc = idx >> 6;
    const int b  = bc / C_;
    const int c  = bc - b * C_;
    float s = 0.f, n = 0.f;
#pragma unroll
    for (int k = 0; k < NCH; ++k) {
        const int blk = b * NCH + k;
        s += psum[blk * (C_ * D_) + c * D_ + d];
        n += pcnt[blk * C_ + c];
    }
    cc[idx] = s / n;
}

// ---------------------------------------------------------------------------
// Kernel 3: the two MLP heads with fp32 WMMA (V_WMMA_F32_16X16X4_F32).
// y = h @ W^T  =>  A = h (16xK tile), B[k][n] = W[n][k].
// A-frag (ISA 7.12.2, 32-bit A 16x4): lane<16 -> {A[l][k0],A[l][k0+1]},
// lanes>=16 -> {A[l-16][k0+2],A[l-16][k0+3]}  => float2 at (l15, 4k+2*hl).
// B-frag mirrors that pattern on rows of W (since B[k][n] = W[n][k]).
// ---------------------------------------------------------------------------
__device__ __forceinline__ v8f gemm64_tile(const float* A,   // 16x64, stride 64
                                           const float* Wn,  // W rows for this N-tile
                                           int l15, int hl) {
    v8f acc = {0.f, 0.f, 0.f, 0.f, 0.f, 0.f, 0.f, 0.f};
#pragma unroll
    for (int kk = 0; kk < 16; ++kk) {
        const int col = kk * 4 + hl * 2;
        v2f a = *(const v2f*)(A  + l15 * 64 + col);
        v2f b = *(const v2f*)(Wn + l15 * 64 + col);
        acc = __builtin_amdgcn_wmma_f32_16x16x4_f32(false, a, false, b,
                                                    (short)0, acc, false, false);
    }
    return acc;
}

__global__ void __launch_bounds__(256)
mlp_kernel(const float* __restrict__ cc,
           const float* __restrict__ wm1, const float* __restrict__ bm1,
           const float* __restrict__ wm2, const float* __restrict__ bm2,
           const float* __restrict__ wm3, const float* __restrict__ bm3,
           const float* __restrict__ wv1, const float* __restrict__ bv1,
           const float* __restrict__ wv2, const float* __restrict__ bv2,
           const float* __restrict__ wv3, const float* __restrict__ bv3,
           float* __restrict__ out) {
    __shared__ float hbuf[2][2][16 * 64];   // [head][double-buffer] 16 KB

    const int tid  = threadIdx.x;
    const int wave = tid >> 5;
    const int lane = tid & 31;
    const int head = wave >> 2;             // 0 = mean, 1 = logvar
    const int nt   = wave & 3;              // N-tile
    const int l15  = lane & 15;
    const int hl   = lane >> 4;
    const int m0   = blockIdx.x * 16;       // slot rows for this block

    const float* W1 = head ? wv1 : wm1;  const float* B1 = head ? bv1 : bm1;
    const float* W2 = head ? wv2 : wm2;  const float* B2 = head ? bv2 : bm2;
    const float* W3 = head ? wv3 : wm3;  const float* B3 = head ? bv3 : bm3;

    const int n = nt * 16 + l15;            // output column of this lane

    // Layer 1: slots(global) -> hbuf[head][0]
    {
        v8f acc = gemm64_tile(cc + m0 * 64, W1 + nt * 16 * 64, l15, hl);
        const float bias = B1[n];
#pragma unroll
        for (int r = 0; r < 8; ++r) {
            const int m = r + 8 * hl;
            hbuf[head][0][m * 64 + n] = fmaxf(acc[r] + bias, 0.f);
        }
    }
    __syncthreads();

    // Layer 2: hbuf[head][0] -> hbuf[head][1]
    {
        v8f acc = gemm64_tile(&hbuf[head][0][0], W2 + nt * 16 * 64, l15, hl);
        const float bias = B2[n];
#pragma unroll
        for (int r = 0; r < 8; ++r) {
            const int m = r + 8 * hl;
            hbuf[head][1][m * 64 + n] = fmaxf(acc[r] + bias, 0.f);
        }
    }
    __syncthreads();

    // Layer 3: hbuf[head][1] -> sigmoid*2-1 -> d_out
    {
        v8f acc = gemm64_tile(&hbuf[head][1][0], W3 + nt * 16 * 64, l15, hl);
        const float bias = B3[n];
        float* dst = out + head * (B_ * C_ * D_);
#pragma unroll
        for (int r = 0; r < 8; ++r) {
            const int m = r + 8 * hl;
            const float z = acc[r] + bias;
            const float sg = 1.f / (1.f + __expf(-z));
            dst[(m0 + m) * 64 + n] = 2.f * sg - 1.f;
        }
    }
}

// ---------------------------------------------------------------------------
// Kernel 4: init_mean / init_logvar = column sums of mean/logvar / (B*K).
// ---------------------------------------------------------------------------
__global__ void __launch_bounds__(128)
init_kernel(float* __restrict__ out, const int* __restrict__ Kp) {
    const int t = threadIdx.x;              // 0..127
    const int head = t >> 6;
    const int d = t & 63;
    const float* src = out + head * (B_ * C_ * D_);
    float s = 0.f;
    for (int r = 0; r < B_ * C_; ++r) s += src[r * 64 + d];
    const float denom = (float)B_ * (float)(*Kp);
    out[2 * (B_ * C_ * D_) + head * 64 + d] = s / denom;
}

// ---------------------------------------------------------------------------
extern "C" void kernel_launch(void* const* d_in, const int* in_sizes, int n_in,
                              void* d_out, int out_size, void* d_ws, size_t ws_size,
                              hipStream_t stream) {
    const float* x      = (const float*)d_in[0];
    const int*   labels = (const int*)  d_in[1];
    const int*   Kp     = (const int*)  d_in[2];
    const float* wm1 = (const float*)d_in[3];  const float* bm1 = (const float*)d_in[4];
    const float* wm2 = (const float*)d_in[5];  const float* bm2 = (const float*)d_in[6];
    const float* wm3 = (const float*)d_in[7];  const float* bm3 = (const float*)d_in[8];
    const float* wv1 = (const float*)d_in[9];  const float* bv1 = (const float*)d_in[10];
    const float* wv2 = (const float*)d_in[11]; const float* bv2 = (const float*)d_in[12];
    const float* wv3 = (const float*)d_in[13]; const float* bv3 = (const float*)d_in[14];
    float* out = (float*)d_out;

    float* ws   = (float*)d_ws;
    float* psum = ws;                                // 512*640
    float* pcnt = psum + (size_t)NBLK * C_ * D_;     // 512*10
    float* cc   = pcnt + (size_t)NBLK * C_;          // 320*64

    segsum_kernel<<<dim3(NCH, B_), 256, 0, stream>>>(x, labels, psum, pcnt);
    reduce_cc_kernel<<<(B_ * C_ * D_ + 255) / 256, 256, 0, stream>>>(psum, pcnt, cc);
    mlp_kernel<<<(B_ * C_) / 16, 256, 0, stream>>>(cc,
        wm1, bm1, wm2, bm2, wm3, bm3, wv1, bv1, wv2, bv2, wv3, bv3, out);
    init_kernel<<<1, 128, 0, stream>>>(out, Kp);
}